// LUTBlock_36601711296516
// MI455X (gfx1250) — compile-verified
//
#include <hip/hip_runtime.h>

#ifndef __has_builtin
#define __has_builtin(x) 0
#endif

#if __has_builtin(__builtin_amdgcn_global_load_async_to_lds_b128) && \
    __has_builtin(__builtin_amdgcn_s_wait_asynccnt)
#define HAS_ASYNC_LDS 1
#else
#define HAS_ASYNC_LDS 0
#endif

#define GLOBAL_AS __attribute__((address_space(1)))
#define LDS_AS    __attribute__((address_space(3)))

typedef int v4i __attribute__((vector_size(16)));   // matches builtin's pointee type

constexpr int kB    = 16384;
constexpr int kInF  = 1024;
constexpr int kOutF = 1024;
constexpr int kT    = 16;
constexpr int kC    = 8;
constexpr int kR    = 256;

#if HAS_ASYNC_LDS
// Two-step casts: addrspacecast first (keeps pointee void), then pointee cast.
__device__ __forceinline__ GLOBAL_AS v4i* as_global_v4i(const void* p) {
    return (GLOBAL_AS v4i*)(GLOBAL_AS void*)p;
}
__device__ __forceinline__ LDS_AS v4i* as_lds_v4i(void* p) {
    return (LDS_AS v4i*)(LDS_AS void*)p;
}
#endif

// One workgroup (256 threads = 8 waves) per batch row.
//   Phase 1: lanes 0..15 each build one 8-bit hard index (16 scalar x-gathers).
//   Phase 2: async-copy the 16 selected 4KB table rows into LDS
//            (global_load_async_to_lds_b128, ASYNCcnt), then each thread
//            reduces its 4 output columns from LDS with ds_load_b128.
__global__ void __launch_bounds__(256)
lut_hard_sum_kernel(const float* __restrict__ x,
                    const float* __restrict__ table,
                    const int*   __restrict__ aa,
                    const int*   __restrict__ ab,
                    float*       __restrict__ out)
{
    const int b   = blockIdx.x;
    const int tid = threadIdx.x;

    __shared__ int s_idx[kT];
#if HAS_ASYNC_LDS
    __shared__ __align__(16) float s_rows[kT * kOutF];   // 64 KB staged gather
#endif

    // ---------------- Phase 1: hard indices ----------------
    if (tid < kT) {
        const float* xr = x + (size_t)b * kInF;
        int idx = 0;
#pragma unroll
        for (int c = 0; c < kC; ++c) {
            const float d = xr[aa[tid * kC + c]] - xr[ab[tid * kC + c]];
            idx |= (d > 0.0f ? 1 : 0) << c;
        }
        s_idx[tid] = idx;
    }
    __syncthreads();

    const int col = tid << 2;                            // 4 floats (16B) / thread

#if HAS_ASYNC_LDS
    // ---------------- Phase 2a: async gather of 16 rows into LDS ----------------
    {
#pragma unroll
        for (int t = 0; t < kT; ++t) {
            const float* src = table + ((size_t)(t * kR + s_idx[t]) * kOutF + col);
            __builtin_amdgcn_global_load_async_to_lds_b128(
                as_global_v4i(src),
                as_lds_v4i(&s_rows[t * kOutF + col]),
                /*imm offset*/ 0, /*cpol*/ 0);
        }
        __builtin_amdgcn_s_wait_asynccnt(0);
    }
    __syncthreads();

    // ---------------- Phase 2b: reduce from LDS ----------------
    float4 acc = {0.0f, 0.0f, 0.0f, 0.0f};
#pragma unroll
    for (int t = 0; t < kT; ++t) {
        const float4 v = *(const float4*)&s_rows[t * kOutF + col];
        acc.x += v.x; acc.y += v.y; acc.z += v.z; acc.w += v.w;
    }
#else
    // Fallback: direct L2-resident gather-accumulate (global_load_b128).
    float4 acc = {0.0f, 0.0f, 0.0f, 0.0f};
#pragma unroll
    for (int t = 0; t < kT; ++t) {
        const float4 v = *(const float4*)(table + ((size_t)(t * kR + s_idx[t]) * kOutF + col));
        acc.x += v.x; acc.y += v.y; acc.z += v.z; acc.w += v.w;
    }
#endif

    *(float4*)(out + (size_t)b * kOutF + col) = acc;
}

extern "C" void kernel_launch(void* const* d_in, const int* in_sizes, int n_in,
                              void* d_out, int out_size, void* d_ws, size_t ws_size,
                              hipStream_t stream)
{
    (void)in_sizes; (void)n_in; (void)out_size; (void)d_ws; (void)ws_size;
    const float* x     = (const float*)d_in[0];
    const float* table = (const float*)d_in[1];
    const int*   aa    = (const int*)d_in[2];
    const int*   ab    = (const int*)d_in[3];
    float*       out   = (float*)d_out;

    dim3 grid(kB);
    dim3 block(256);
    lut_hard_sum_kernel<<<grid, block, 0, stream>>>(x, table, aa, ab, out);
}